// MatrixGraphConvolution_79422535238375
// MI455X (gfx1250) — compile-verified
//
#include <hip/hip_runtime.h>

typedef __attribute__((ext_vector_type(2))) float v2f;
typedef __attribute__((ext_vector_type(8))) float v8f;

#define F 128  // F_IN == F_OUT == 128 per reference

// ---------------------------------------------------------------------------
// Kernel 1: zero mask + deg region of workspace (harness poisons ws to 0xAA)
// ---------------------------------------------------------------------------
__global__ void gc_zero_kernel(unsigned* __restrict__ p, int n) {
    int i = blockIdx.x * blockDim.x + threadIdx.x;
    int stride = gridDim.x * blockDim.x;
    for (; i < n; i += stride) p[i] = 0u;
}

// ---------------------------------------------------------------------------
// Kernel 2: build bitmask adjacency (dedup via atomicOr) + duplicate-counting
// in-degree (atomicAdd), matching reference set()/bincount semantics.
// ---------------------------------------------------------------------------
__global__ void gc_build_adj_kernel(const int* __restrict__ src,
                                    const int* __restrict__ dst,
                                    int nE,
                                    unsigned* __restrict__ mask,
                                    unsigned* __restrict__ deg,
                                    int wpr) {
    int e = blockIdx.x * blockDim.x + threadIdx.x;
    if (e >= nE) return;
    int s = src[e];
    int d = dst[e];
    atomicAdd(&deg[d], 1u);
    atomicOr(&mask[(size_t)d * wpr + (s >> 5)], 1u << (s & 31));
}

// ---------------------------------------------------------------------------
// Kernel 3: h = x @ W^T  (to ws) and out = x @ B^T (to d_out), fused so the
// A operand (x tile) is loaded once. One wave per 16x16 tile; 8 waves/block
// cover the 128 output features; blockIdx.x covers 16-row tiles.
// V_WMMA_F32_16X16X4_F32 layout: A 16x4 -> lane&15 = M, half-wave = K pair.
// ---------------------------------------------------------------------------
__global__ void gc_xw_xb_kernel(const float* __restrict__ x,
                                const float* __restrict__ W,
                                const float* __restrict__ B,
                                float* __restrict__ h,
                                float* __restrict__ out) {
    const int lane = threadIdx.x & 31;
    const int wave = threadIdx.x >> 5;
    const int r0   = blockIdx.x * 16;
    const int c0   = wave * 16;
    const int m    = lane & 15;           // M for A, N for B/C
    const int koff = (lane >> 4) << 1;    // half-wave K offset (0 or 2)

    const float* xrow = x + (size_t)(r0 + m) * F;   // A rows
    const float* wrow = W + (size_t)(c0 + m) * F;   // B[k][n] = W[n][k]
    const float* brow = B + (size_t)(c0 + m) * F;

    v8f ch = {};
    v8f cg = {};
    for (int k0 = 0; k0 < F; k0 += 4) {
        v2f a  = { xrow[k0 + koff], xrow[k0 + koff + 1] };
        v2f bw = { wrow[k0 + koff], wrow[k0 + koff + 1] };
        v2f bb = { brow[k0 + koff], brow[k0 + koff + 1] };
        ch = __builtin_amdgcn_wmma_f32_16x16x4_f32(false, a, false, bw,
                                                   (short)0, ch, false, false);
        cg = __builtin_amdgcn_wmma_f32_16x16x4_f32(false, a, false, bb,
                                                   (short)0, cg, false, false);
    }

    const int hi  = lane >> 4;
    const int col = c0 + m;
#pragma unroll
    for (int r = 0; r < 8; ++r) {
        int row = r0 + r + 8 * hi;        // C/D layout: VGPR r -> M = r (+8 hi half)
        h  [(size_t)row * F + col] = ch[r];
        out[(size_t)row * F + col] = cg[r];
    }
}

// ---------------------------------------------------------------------------
// Kernel 4: out += (A @ h) / deg. A expanded on the fly from the bitmask into
// exact {0.0,1.0} f32 WMMA operands.
//
// Skip logic: OR the mask word across the 16 tile rows with a 4-step
// ds_swizzle XOR reduction (half-waves hold duplicate rows, so xor-16 is not
// needed), broadcast to an SGPR with readfirstlane, then guard each 16x16x4
// WMMA K-step with a *scalar* nibble test. Branches are s_cbranch on SCC:
// EXEC stays all-ones around every WMMA by construction, and ~78% of K-steps
// (and their h loads) are skipped at this edge density.
// ---------------------------------------------------------------------------
__global__ void gc_aggregate_kernel(const unsigned* __restrict__ mask,
                                    const unsigned* __restrict__ deg,
                                    const float* __restrict__ h,
                                    float* __restrict__ out,
                                    int wpr) {
    const int lane = threadIdx.x & 31;
    const int wave = threadIdx.x >> 5;
    const int r0   = blockIdx.x * 16;
    const int c0   = wave * 16;
    const int m    = lane & 15;
    const int koff = (lane >> 4) << 1;

    const unsigned* mrow = mask + (size_t)(r0 + m) * wpr;
    v8f acc = {};

    for (int kw = 0; kw < wpr; ++kw) {
        unsigned word = mrow[kw];
        // OR across the 16 rows of this tile (lanes 0-15 / 16-31 duplicate)
        unsigned ow = word;
        ow |= (unsigned)__builtin_amdgcn_ds_swizzle((int)ow, 0x041F); // xor 1
        ow |= (unsigned)__builtin_amdgcn_ds_swizzle((int)ow, 0x081F); // xor 2
        ow |= (unsigned)__builtin_amdgcn_ds_swizzle((int)ow, 0x101F); // xor 4
        ow |= (unsigned)__builtin_amdgcn_ds_swizzle((int)ow, 0x201F); // xor 8
        unsigned sw = (unsigned)__builtin_amdgcn_readfirstlane((int)ow);
        if (sw == 0u) continue;                       // scalar whole-word skip

        const int kbase = kw << 5;
        const float* hk = h + (size_t)kbase * F + c0 + m;  // B[k][n] = h[k][col]
#pragma unroll
        for (int s = 0; s < 8; ++s) {
            if (((sw >> (s << 2)) & 0xFu) == 0u) continue; // scalar nibble skip
            const int kl = (s << 2) + koff;
            v2f a = { (float)((word >> kl) & 1u),
                      (float)((word >> (kl + 1)) & 1u) };
            v2f b = { hk[(size_t)kl * F], hk[(size_t)(kl + 1) * F] };
            acc = __builtin_amdgcn_wmma_f32_16x16x4_f32(false, a, false, b,
                                                        (short)0, acc, false, false);
        }
    }

    const int hi  = lane >> 4;
    const int col = c0 + m;
#pragma unroll
    for (int r = 0; r < 8; ++r) {
        int row = r0 + r + 8 * hi;
        float d = (float)deg[row];
        if (d == 0.0f) d = 1.0f;          // degree-0 -> 1, per reference
        out[(size_t)row * F + col] += acc[r] / d;
    }
}

// ---------------------------------------------------------------------------
extern "C" void kernel_launch(void* const* d_in, const int* in_sizes, int n_in,
                              void* d_out, int out_size, void* d_ws, size_t ws_size,
                              hipStream_t stream) {
    const float* x    = (const float*)d_in[0];
    const int*   edge = (const int*)  d_in[1];   // [2, E]: row 0 = src, row 1 = dst
    const float* W    = (const float*)d_in[2];
    const float* B    = (const float*)d_in[3];
    float*       out  = (float*)d_out;

    const int N   = in_sizes[0] / F;     // 8192 nodes
    const int E   = in_sizes[1] / 2;     // 262144 edges
    const int wpr = (N + 31) / 32;       // mask words per row (256)

    // workspace layout: [ mask: N*wpr u32 | deg: N u32 | h: N*F f32 ]
    unsigned* mask = (unsigned*)d_ws;
    unsigned* deg  = mask + (size_t)N * wpr;
    float*    h    = (float*)(deg + N);

    const int*   src = edge;
    const int*   dst = edge + E;

    // 1) zero mask + deg
    int zn = N * wpr + N;
    int zgrid = (zn + 255) / 256;
    if (zgrid > 4096) zgrid = 4096;
    gc_zero_kernel<<<zgrid, 256, 0, stream>>>(mask, zn);

    // 2) scatter edges into bitmask + degree
    gc_build_adj_kernel<<<(E + 255) / 256, 256, 0, stream>>>(src, dst, E, mask, deg, wpr);

    // 3) h = x@W^T (ws), out = x@B^T (d_out)   [512 blocks x 8 waves]
    gc_xw_xb_kernel<<<N / 16, 256, 0, stream>>>(x, W, B, h, out);

    // 4) out += (A @ h) / deg
    gc_aggregate_kernel<<<N / 16, 256, 0, stream>>>(mask, deg, h, out, wpr);
}